// VNSmall_35012573397750
// MI455X (gfx1250) — compile-verified
//
#include <hip/hip_runtime.h>
#include <math.h>

typedef float v2f __attribute__((ext_vector_type(2)));
typedef float v8f __attribute__((ext_vector_type(8)));

#define BB   8
#define NN   4096
#define KNN  20
#define CC   21
#define CC2  4
#define EPSF 1e-6f
#define BNEPS 1e-5f
#define TPAD 17   // LDS tile row stride (floats): 17 coprime with 64 banks

// ---------------------------------------------------------------------------
// Wave (32-lane) sum reduction
// ---------------------------------------------------------------------------
__device__ __forceinline__ float wred(float v) {
#pragma unroll
    for (int off = 16; off > 0; off >>= 1) v += __shfl_down(v, off, 32);
    return v;
}

// ---------------------------------------------------------------------------
// Kernel 0: zero stats blocks + output accumulator (determinism per call)
// ---------------------------------------------------------------------------
__global__ void k_init(float* s0, float* s1, float* sb, float* s2, float* out) {
    int t = threadIdx.x;
    if (t < 2 * CC) { s0[t] = 0.f; s1[t] = 0.f; sb[t] = 0.f; }
    if (t < 2 * CC2) s2[t] = 0.f;
    if (t < BB * 9) out[t] = 0.f;
}

// ---------------------------------------------------------------------------
// Kernel 1: pairwise neg-distance (WMMA f32 16x16x4) + per-row top-20
// One wave handles 16 rows of one batch; sweeps 256 column tiles.
//   neg_dist = 2 p.q - |p|^2 - |q|^2  ->  A = 2p (16x4), B = q (4x16),
//   C preloaded with (-|p_row|^2) - |q_col|^2; one V_WMMA per tile.
// ---------------------------------------------------------------------------
__global__ __launch_bounds__(32)
void k_topk(const float* __restrict__ pc, int* __restrict__ idx_out) {
    __shared__ float s_xxrow[16];
    __shared__ float s_tile[16 * TPAD];
    __shared__ float s_mval[16 * 2 * KNN];
    __shared__ int   s_midx[16 * 2 * KNN];

    const int l      = threadIdx.x;      // 0..31
    const int b      = blockIdx.x >> 8;  // 256 row tiles per batch
    const int m0     = (blockIdx.x & 255) * 16;
    const int lane16 = l & 15;
    const int hi     = l >> 4;

    const float* P = pc + (size_t)b * 3 * NN;

    // row point (A fragment source): lane holds M = lane16, K = hi?{2,3}:{0,1}
    const int m = m0 + lane16;
    float rx = P[0 * NN + m], ry = P[1 * NN + m], rz = P[2 * NN + m];
    if (hi == 0) s_xxrow[lane16] = rx * rx + ry * ry + rz * rz;
    __syncthreads();

    float nrowxx[8];   // hoisted negation: fold into subtract in the loop
#pragma unroll
    for (int v = 0; v < 8; ++v) nrowxx[v] = -s_xxrow[v + 8 * hi];

    v2f afrag;
    afrag.x = 2.0f * (hi ? rz : rx);   // K = 2 (z) | K = 0 (x)
    afrag.y = 2.0f * (hi ? 0.f : ry);  // K = 3 pad | K = 1 (y)

    float bestv[KNN];
    int   besti[KNN];
#pragma unroll
    for (int i = 0; i < KNN; ++i) { bestv[i] = -3.0e38f; besti[i] = 0; }

    const int r = l >> 1;   // scan row for this lane
    const int h = l & 1;    // column half

    for (int t = 0; t < NN / 16; ++t) {
        const int n0 = t * 16;
        const int n  = n0 + lane16;
        float cx = P[0 * NN + n], cy = P[1 * NN + n], cz = P[2 * NN + n];
        float cxx = cx * cx + cy * cy + cz * cz;

        v2f bfrag;
        bfrag.x = hi ? cz : cx;
        bfrag.y = hi ? 0.f : cy;

        v8f cacc;
#pragma unroll
        for (int v = 0; v < 8; ++v) cacc[v] = nrowxx[v] - cxx;

        v8f dacc = __builtin_amdgcn_wmma_f32_16x16x4_f32(
            false, afrag, false, bfrag, (short)0, cacc, false, false);

        // C/D layout: lane holds col N=lane16, rows M = v + 8*hi
#pragma unroll
        for (int v = 0; v < 8; ++v)
            s_tile[(v + 8 * hi) * TPAD + lane16] = dacc[v];
        __syncthreads();

        // batch the 8 candidate loads (one dscnt wait, conflict-free banks)
        float cand[8];
#pragma unroll
        for (int j = 0; j < 8; ++j)
            cand[j] = s_tile[r * TPAD + h * 8 + j];

        // top-k maintenance: 2 lanes per row, 8 columns each
#pragma unroll
        for (int j = 0; j < 8; ++j) {
            float val = cand[j];
            int   gi  = n0 + h * 8 + j;
            if (val > bestv[KNN - 1]) {
                bestv[KNN - 1] = val; besti[KNN - 1] = gi;
#pragma unroll
                for (int s = KNN - 1; s > 0; --s) {
                    if (bestv[s] > bestv[s - 1]) {
                        float tv = bestv[s]; bestv[s] = bestv[s - 1]; bestv[s - 1] = tv;
                        int   ti = besti[s]; besti[s] = besti[s - 1]; besti[s - 1] = ti;
                    }
                }
            }
        }
        __syncthreads();
    }

    // merge the two per-row lists
#pragma unroll
    for (int i = 0; i < KNN; ++i) {
        s_mval[(r * 2 + h) * KNN + i] = bestv[i];
        s_midx[(r * 2 + h) * KNN + i] = besti[i];
    }
    __syncthreads();

    if (hi == 0) {
        const int rr = lane16;
        const float* va = &s_mval[(rr * 2 + 0) * KNN];
        const int*   ia = &s_midx[(rr * 2 + 0) * KNN];
        const float* vb = &s_mval[(rr * 2 + 1) * KNN];
        const int*   ib = &s_midx[(rr * 2 + 1) * KNN];
        int pa = 0, pb = 0;
        int* outp = idx_out + ((size_t)b * NN + (m0 + rr)) * KNN;
        for (int i = 0; i < KNN; ++i) {
            bool takeA;
            if (pa >= KNN)            takeA = false;
            else if (pb >= KNN)       takeA = true;
            else if (va[pa] > vb[pb]) takeA = true;
            else if (va[pa] < vb[pb]) takeA = false;
            else                      takeA = (ia[pa] < ib[pb]);
            outp[i] = takeA ? ia[pa++] : ib[pb++];
        }
    }
}

// ---------------------------------------------------------------------------
// Graph-feature vectors for (center n, neighbor j): {nbr-ctr, ctr, nbr x ctr}
// ---------------------------------------------------------------------------
__device__ __forceinline__ void vn_feat(const float* __restrict__ P, int n, int j,
                                        float v[3][3]) {
    float cx = P[n],      cy = P[NN + n],  cz = P[2 * NN + n];
    float nx = P[j],      ny = P[NN + j],  nz = P[2 * NN + j];
    v[0][0] = nx - cx; v[0][1] = ny - cy; v[0][2] = nz - cz;
    v[1][0] = cx;      v[1][1] = cy;      v[1][2] = cz;
    v[2][0] = ny * cz - nz * cy;
    v[2][1] = nz * cx - nx * cz;
    v[2][2] = nx * cy - ny * cx;
}

// ---------------------------------------------------------------------------
// Kernel 2: norm statistics of p = Wp_f @ feat over (b,n,k) per channel
// ---------------------------------------------------------------------------
__global__ __launch_bounds__(256)
void k_stats0(const float* __restrict__ pc, const int* __restrict__ idx,
              const float* __restrict__ Wf, float* __restrict__ stats) {
    float s1[CC], s2[CC];
#pragma unroll
    for (int o = 0; o < CC; ++o) { s1[o] = 0.f; s2[o] = 0.f; }

    const int total = BB * NN * KNN;
    for (int it = blockIdx.x * blockDim.x + threadIdx.x; it < total;
         it += gridDim.x * blockDim.x) {
        int n = (it / KNN) % NN;
        int b = it / (KNN * NN);
        const float* P = pc + (size_t)b * 3 * NN;
        int j = idx[it];
        float v[3][3];
        vn_feat(P, n, j, v);
#pragma unroll
        for (int o = 0; o < CC; ++o) {
            float w0 = Wf[o * 3], w1 = Wf[o * 3 + 1], w2 = Wf[o * 3 + 2];
            float px = w0 * v[0][0] + w1 * v[1][0] + w2 * v[2][0];
            float py = w0 * v[0][1] + w1 * v[1][1] + w2 * v[2][1];
            float pz = w0 * v[0][2] + w1 * v[1][2] + w2 * v[2][2];
            float nrm = sqrtf(px * px + py * py + pz * pz) + EPSF;
            s1[o] += nrm; s2[o] += nrm * nrm;
        }
    }
#pragma unroll
    for (int o = 0; o < CC; ++o) {
        float a = wred(s1[o]);
        float c = wred(s2[o]);
        if ((threadIdx.x & 31) == 0) {
            atomicAdd(&stats[o], a);
            atomicAdd(&stats[CC + o], c);
        }
    }
}

// ---------------------------------------------------------------------------
// Kernel 3: VN-linear-relu layer 1 + mean over K  -> x1[b][o][d][n]
// ---------------------------------------------------------------------------
__global__ __launch_bounds__(128)
void k_layer1(const float* __restrict__ pc, const int* __restrict__ idx,
              const float* __restrict__ Wf, const float* __restrict__ Wd,
              const float* __restrict__ g, const float* __restrict__ be,
              const float* __restrict__ stats, float* __restrict__ x1) {
    int tid = blockIdx.x * blockDim.x + threadIdx.x;
    if (tid >= BB * NN) return;
    int n = tid % NN, b = tid / NN;
    const float* P = pc + (size_t)b * 3 * NN;

    float acc[CC][3];
#pragma unroll
    for (int o = 0; o < CC; ++o) { acc[o][0] = 0.f; acc[o][1] = 0.f; acc[o][2] = 0.f; }

    const float invcnt = 1.0f / (float)(BB * NN * KNN);
    for (int k = 0; k < KNN; ++k) {
        int j = idx[(size_t)tid * KNN + k];
        float v[3][3];
        vn_feat(P, n, j, v);
#pragma unroll
        for (int o = 0; o < CC; ++o) {
            float wf0 = Wf[o * 3], wf1 = Wf[o * 3 + 1], wf2 = Wf[o * 3 + 2];
            float wd0 = Wd[o * 3], wd1 = Wd[o * 3 + 1], wd2 = Wd[o * 3 + 2];
            float px = wf0 * v[0][0] + wf1 * v[1][0] + wf2 * v[2][0];
            float py = wf0 * v[0][1] + wf1 * v[1][1] + wf2 * v[2][1];
            float pz = wf0 * v[0][2] + wf1 * v[1][2] + wf2 * v[2][2];
            float dx = wd0 * v[0][0] + wd1 * v[1][0] + wd2 * v[2][0];
            float dy = wd0 * v[0][1] + wd1 * v[1][1] + wd2 * v[2][1];
            float dz = wd0 * v[0][2] + wd1 * v[1][2] + wd2 * v[2][2];
            float nrm = sqrtf(px * px + py * py + pz * pz) + EPSF;
            float mu  = stats[o] * invcnt;
            float var = stats[CC + o] * invcnt - mu * mu;
            float nbn = (nrm - mu) * rsqrtf(var + BNEPS) * g[o] + be[o];
            float s = nbn / nrm;
            px *= s; py *= s; pz *= s;
            float dot = px * dx + py * dy + pz * dz;
            if (dot < 0.f) {
                float f = dot / (dx * dx + dy * dy + dz * dz + EPSF);
                px -= f * dx; py -= f * dy; pz -= f * dz;
            }
            acc[o][0] += px; acc[o][1] += py; acc[o][2] += pz;
        }
    }
    const float invK = 1.0f / (float)KNN;
#pragma unroll
    for (int o = 0; o < CC; ++o)
#pragma unroll
        for (int d = 0; d < 3; ++d)
            x1[(((size_t)b * CC + o) * 3 + d) * NN + n] = acc[o][d] * invK;
}

// ---------------------------------------------------------------------------
// Kernel 4: norm statistics of q = W1_f @ x1 over (b,n) per channel
// ---------------------------------------------------------------------------
__global__ __launch_bounds__(128)
void k_stats1(const float* __restrict__ x1, const float* __restrict__ Wf,
              float* __restrict__ stats) {
    int tid = blockIdx.x * blockDim.x + threadIdx.x;
    if (tid >= BB * NN) return;
    int n = tid % NN, b = tid / NN;
    float x[CC][3];
#pragma unroll
    for (int c = 0; c < CC; ++c)
#pragma unroll
        for (int d = 0; d < 3; ++d)
            x[c][d] = x1[(((size_t)b * CC + c) * 3 + d) * NN + n];
#pragma unroll
    for (int o = 0; o < CC; ++o) {
        float qx = 0.f, qy = 0.f, qz = 0.f;
#pragma unroll
        for (int c = 0; c < CC; ++c) {
            float w = Wf[o * CC + c];
            qx += w * x[c][0]; qy += w * x[c][1]; qz += w * x[c][2];
        }
        float nrm = sqrtf(qx * qx + qy * qy + qz * qz) + EPSF;
        float a = wred(nrm);
        float c2 = wred(nrm * nrm);
        if ((threadIdx.x & 31) == 0) {
            atomicAdd(&stats[o], a);
            atomicAdd(&stats[CC + o], c2);
        }
    }
}

// ---------------------------------------------------------------------------
// Kernel 5: VN-linear-relu layer 2 -> y ; fused norm stats of y for vn_bn
// ---------------------------------------------------------------------------
__global__ __launch_bounds__(128)
void k_layer2(const float* __restrict__ x1, const float* __restrict__ Wf,
              const float* __restrict__ Wd, const float* __restrict__ g,
              const float* __restrict__ be, const float* __restrict__ stats,
              float* __restrict__ y, float* __restrict__ statsb) {
    int tid = blockIdx.x * blockDim.x + threadIdx.x;
    if (tid >= BB * NN) return;
    int n = tid % NN, b = tid / NN;
    float x[CC][3];
#pragma unroll
    for (int c = 0; c < CC; ++c)
#pragma unroll
        for (int d = 0; d < 3; ++d)
            x[c][d] = x1[(((size_t)b * CC + c) * 3 + d) * NN + n];

    const float invcnt = 1.0f / (float)(BB * NN);
#pragma unroll
    for (int o = 0; o < CC; ++o) {
        float px = 0.f, py = 0.f, pz = 0.f, dx = 0.f, dy = 0.f, dz = 0.f;
#pragma unroll
        for (int c = 0; c < CC; ++c) {
            float wf = Wf[o * CC + c], wd = Wd[o * CC + c];
            px += wf * x[c][0]; py += wf * x[c][1]; pz += wf * x[c][2];
            dx += wd * x[c][0]; dy += wd * x[c][1]; dz += wd * x[c][2];
        }
        float nrm = sqrtf(px * px + py * py + pz * pz) + EPSF;
        float mu  = stats[o] * invcnt;
        float var = stats[CC + o] * invcnt - mu * mu;
        float nbn = (nrm - mu) * rsqrtf(var + BNEPS) * g[o] + be[o];
        float s = nbn / nrm;
        px *= s; py *= s; pz *= s;
        float dot = px * dx + py * dy + pz * dz;
        if (dot < 0.f) {
            float f = dot / (dx * dx + dy * dy + dz * dz + EPSF);
            px -= f * dx; py -= f * dy; pz -= f * dz;
        }
        y[(((size_t)b * CC + o) * 3 + 0) * NN + n] = px;
        y[(((size_t)b * CC + o) * 3 + 1) * NN + n] = py;
        y[(((size_t)b * CC + o) * 3 + 2) * NN + n] = pz;
        float nrmy = sqrtf(px * px + py * py + pz * pz) + EPSF;
        float a = wred(nrmy);
        float c2 = wred(nrmy * nrmy);
        if ((threadIdx.x & 31) == 0) {
            atomicAdd(&statsb[o], a);
            atomicAdd(&statsb[CC + o], c2);
        }
    }
}

// ---------------------------------------------------------------------------
// z = vn_bn(y) (shared by kernels 6/7)
// ---------------------------------------------------------------------------
__device__ __forceinline__ void load_z(const float* __restrict__ y,
                                       const float* __restrict__ statsb,
                                       const float* __restrict__ g,
                                       const float* __restrict__ be,
                                       int b, int n, float z[CC][3]) {
    const float inv = 1.0f / (float)(BB * NN);
#pragma unroll
    for (int c = 0; c < CC; ++c) {
        float v0 = y[(((size_t)b * CC + c) * 3 + 0) * NN + n];
        float v1 = y[(((size_t)b * CC + c) * 3 + 1) * NN + n];
        float v2 = y[(((size_t)b * CC + c) * 3 + 2) * NN + n];
        float nrm = sqrtf(v0 * v0 + v1 * v1 + v2 * v2) + EPSF;
        float mu  = statsb[c] * inv;
        float var = statsb[CC + c] * inv - mu * mu;
        float nbn = (nrm - mu) * rsqrtf(var + BNEPS) * g[c] + be[c];
        float s = nbn / nrm;
        z[c][0] = v0 * s; z[c][1] = v1 * s; z[c][2] = v2 * s;
    }
}

// ---------------------------------------------------------------------------
// Kernel 6: norm statistics of q2 = W2_f @ z per channel (4) over (b,n)
// ---------------------------------------------------------------------------
__global__ __launch_bounds__(128)
void k_stats2(const float* __restrict__ y, const float* __restrict__ statsb,
              const float* __restrict__ g1, const float* __restrict__ b1,
              const float* __restrict__ Wf, float* __restrict__ stats) {
    int tid = blockIdx.x * blockDim.x + threadIdx.x;
    if (tid >= BB * NN) return;
    int n = tid % NN, b = tid / NN;
    float z[CC][3];
    load_z(y, statsb, g1, b1, b, n, z);
#pragma unroll
    for (int o = 0; o < CC2; ++o) {
        float qx = 0.f, qy = 0.f, qz = 0.f;
#pragma unroll
        for (int c = 0; c < CC; ++c) {
            float w = Wf[o * CC + c];
            qx += w * z[c][0]; qy += w * z[c][1]; qz += w * z[c][2];
        }
        float nrm = sqrtf(qx * qx + qy * qy + qz * qz) + EPSF;
        float a = wred(nrm);
        float c2 = wred(nrm * nrm);
        if ((threadIdx.x & 31) == 0) {
            atomicAdd(&stats[o], a);
            atomicAdd(&stats[CC2 + o], c2);
        }
    }
}

// ---------------------------------------------------------------------------
// Kernel 7: layer 3 VN-linear-relu + mean over N -> out[b][o<3][d]
// ---------------------------------------------------------------------------
__global__ __launch_bounds__(128)
void k_final(const float* __restrict__ y, const float* __restrict__ statsb,
             const float* __restrict__ g1, const float* __restrict__ b1,
             const float* __restrict__ Wf, const float* __restrict__ Wd,
             const float* __restrict__ stats2, const float* __restrict__ g2,
             const float* __restrict__ b2, float* __restrict__ out) {
    int tid = blockIdx.x * blockDim.x + threadIdx.x;
    if (tid >= BB * NN) return;
    int n = tid % NN, b = tid / NN;
    float z[CC][3];
    load_z(y, statsb, g1, b1, b, n, z);

    float res[3][3];
    const float invcnt = 1.0f / (float)(BB * NN);
#pragma unroll
    for (int o = 0; o < CC2; ++o) {
        float px = 0.f, py = 0.f, pz = 0.f, dx = 0.f, dy = 0.f, dz = 0.f;
#pragma unroll
        for (int c = 0; c < CC; ++c) {
            float wf = Wf[o * CC + c], wd = Wd[o * CC + c];
            px += wf * z[c][0]; py += wf * z[c][1]; pz += wf * z[c][2];
            dx += wd * z[c][0]; dy += wd * z[c][1]; dz += wd * z[c][2];
        }
        float nrm = sqrtf(px * px + py * py + pz * pz) + EPSF;
        float mu  = stats2[o] * invcnt;
        float var = stats2[CC2 + o] * invcnt - mu * mu;
        float nbn = (nrm - mu) * rsqrtf(var + BNEPS) * g2[o] + b2[o];
        float s = nbn / nrm;
        px *= s; py *= s; pz *= s;
        float dot = px * dx + py * dy + pz * dz;
        if (dot < 0.f) {
            float f = dot / (dx * dx + dy * dy + dz * dz + EPSF);
            px -= f * dx; py -= f * dy; pz -= f * dz;
        }
        if (o < 3) { res[o][0] = px; res[o][1] = py; res[o][2] = pz; }
    }
    const float invN = 1.0f / (float)NN;
#pragma unroll
    for (int o = 0; o < 3; ++o)
#pragma unroll
        for (int d = 0; d < 3; ++d) {
            float v = wred(res[o][d] * invN);
            if ((threadIdx.x & 31) == 0)
                atomicAdd(&out[b * 9 + o * 3 + d], v);
        }
}

// ---------------------------------------------------------------------------
// Launch
// ---------------------------------------------------------------------------
extern "C" void kernel_launch(void* const* d_in, const int* in_sizes, int n_in,
                              void* d_out, int out_size, void* d_ws, size_t ws_size,
                              hipStream_t stream) {
    const float* pc    = (const float*)d_in[0];
    const float* Wp_f  = (const float*)d_in[1];
    const float* Wp_d  = (const float*)d_in[2];
    const float* bnp_g = (const float*)d_in[3];
    const float* bnp_b = (const float*)d_in[4];
    const float* W1_f  = (const float*)d_in[5];
    const float* W1_d  = (const float*)d_in[6];
    const float* bn1i_g= (const float*)d_in[7];
    const float* bn1i_b= (const float*)d_in[8];
    const float* bn1_g = (const float*)d_in[9];
    const float* bn1_b = (const float*)d_in[10];
    const float* W2_f  = (const float*)d_in[11];
    const float* W2_d  = (const float*)d_in[12];
    const float* bn2_g = (const float*)d_in[13];
    const float* bn2_b = (const float*)d_in[14];
    float* out = (float*)d_out;

    char* ws = (char*)d_ws;
    size_t off = 0;
    int* idx = (int*)(ws + off);          off += (size_t)BB * NN * KNN * sizeof(int);
    float* stats0 = (float*)(ws + off);   off += 64 * sizeof(float);
    float* stats1 = (float*)(ws + off);   off += 64 * sizeof(float);
    float* statsb = (float*)(ws + off);   off += 64 * sizeof(float);
    float* stats2 = (float*)(ws + off);   off += 64 * sizeof(float);
    float* x1 = (float*)(ws + off);       off += (size_t)BB * CC * 3 * NN * sizeof(float);
    float* yb = (float*)(ws + off);       off += (size_t)BB * CC * 3 * NN * sizeof(float);

    const int pnBlocks = (BB * NN) / 128;   // 256

    k_init<<<1, 128, 0, stream>>>(stats0, stats1, statsb, stats2, out);
    k_topk<<<BB * (NN / 16), 32, 0, stream>>>(pc, idx);
    k_stats0<<<64, 256, 0, stream>>>(pc, idx, Wp_f, stats0);
    k_layer1<<<pnBlocks, 128, 0, stream>>>(pc, idx, Wp_f, Wp_d, bnp_g, bnp_b, stats0, x1);
    k_stats1<<<pnBlocks, 128, 0, stream>>>(x1, W1_f, stats1);
    k_layer2<<<pnBlocks, 128, 0, stream>>>(x1, W1_f, W1_d, bn1i_g, bn1i_b, stats1, yb, statsb);
    k_stats2<<<pnBlocks, 128, 0, stream>>>(yb, statsb, bn1_g, bn1_b, W2_f, stats2);
    k_final<<<pnBlocks, 128, 0, stream>>>(yb, statsb, bn1_g, bn1_b, W2_f, W2_d,
                                          stats2, bn2_g, bn2_b, out);
}